// DigitCaps_10977936408608
// MI455X (gfx1250) — compile-verified
//
#include <hip/hip_runtime.h>
#include <hip/hip_bf16.h>

typedef __attribute__((ext_vector_type(2))) float v2f;
typedef __attribute__((ext_vector_type(8))) float v8f;

#define B_TOT 512
#define C_IN  1152
#define O_CAPS 10
#define D_DIM 16
#define I_IN  8

// ---------------------------------------------------------------------------
// Transpose x[b][c][i] -> xT[(c*8+i)][b]  (coalesced A-operand loads later)
// ---------------------------------------------------------------------------
__global__ __launch_bounds__(256) void transpose_x_kernel(
    const float* __restrict__ x, float* __restrict__ xT)
{
    int t = blockIdx.x * 256 + threadIdx.x;        // [0, B*C*I)
    int k = t % (C_IN * I_IN);
    int b = t / (C_IN * I_IN);
    xT[k * B_TOT + b] = x[t];
}

// ---------------------------------------------------------------------------
// s[b,o,d] = sum_{c,i} (coeff[b,o,c] * x[b,c,i]) * W[o,c,d,i]
// One WMMA tile: M = 16 batches, N = 16 out-dims, K accumulated over c,i.
// MODE 0: coeff = 0.1 uniform, write s[O][B][D]
// MODE 1: coeff from scratch,  write s[O][B][D]
// MODE 2: coeff from scratch,  write out[B][O][D]  (final v)
// ---------------------------------------------------------------------------
template <int MODE>
__global__ __launch_bounds__(256) void gemm_s_kernel(
    const float* __restrict__ xT,      // [C*I][B]
    const float* __restrict__ W,       // [O][C][D][I]
    const float* __restrict__ coeff,   // [O][C][B]
    float* __restrict__ outp)
{
    const int tid  = threadIdx.x;
    const int lane = tid & 31;
    const int wave = tid >> 5;
    const int o    = blockIdx.x;       // 0..9
    const int bb   = blockIdx.y * 16;  // batch tile base
    const int bl   = lane & 15;        // batch row (M) within tile
    const int hi   = lane >> 4;        // K sub-block select (ISA A/B layout)
    const int ia   = hi * 2;

    __shared__ float red[256 * 8];

    v8f acc = {};
    const int c0 = wave * (C_IN / 8);

    const float* xp = xT + (size_t)(c0 * I_IN + ia) * B_TOT + bb + bl;
    const float* wp = W + ((size_t)(o * C_IN + c0) * D_DIM + bl) * I_IN + ia;
    const float* cp = coeff + (size_t)(o * C_IN + c0) * B_TOT + bb + bl;

#pragma unroll 2
    for (int c = 0; c < C_IN / 8; ++c) {
        float cf = (MODE == 0) ? 0.1f : cp[0];
        v2f a, bm;
        // K-step 0: i = ia, ia+1
        a.x = xp[0 * B_TOT] * cf;
        a.y = xp[1 * B_TOT] * cf;
        bm  = *(const v2f*)(wp);
        acc = __builtin_amdgcn_wmma_f32_16x16x4_f32(false, a, false, bm,
                                                    (short)0, acc, false, false);
        // K-step 1: i = ia+4, ia+5
        a.x = xp[4 * B_TOT] * cf;
        a.y = xp[5 * B_TOT] * cf;
        bm  = *(const v2f*)(wp + 4);
        acc = __builtin_amdgcn_wmma_f32_16x16x4_f32(false, a, false, bm,
                                                    (short)0, acc, false, false);
        xp += I_IN * B_TOT;
        wp += D_DIM * I_IN;
        cp += B_TOT;
    }

    // split-K reduction across the 8 waves via LDS
#pragma unroll
    for (int j = 0; j < 8; ++j) red[tid * 8 + j] = acc[j];
    __syncthreads();
    if (wave == 0) {
#pragma unroll
        for (int j = 0; j < 8; ++j) {
            float s = 0.f;
#pragma unroll
            for (int w = 0; w < 8; ++w) s += red[(w * 32 + lane) * 8 + j];
            int bg = bb + ((lane < 16) ? j : j + 8);   // C/D tile: M row
            int d  = lane & 15;                        // N col
            if (MODE == 2) outp[(bg * O_CAPS + o) * D_DIM + d] = s;
            else           outp[(o * B_TOT + bg) * D_DIM + d] = s;
        }
    }
}

// ---------------------------------------------------------------------------
// logits[b,o,c] (+)= uv = sum_d u_hat[b,o,c,d]*s[b,o,d]
//   via g[b,(c,i)] = sum_d s[b,d]*W[o,c,d,i]  (WMMA, K=16 over d)
//   then uv[b,c]   = sum_i x[b,c,i]*g[b,c,i]  (lane-group reduce)
// ---------------------------------------------------------------------------
template <bool ACC>
__global__ __launch_bounds__(256) void gemm_uv_kernel(
    const float* __restrict__ x,       // [B][C][I]
    const float* __restrict__ W,       // [O][C][D][I]
    const float* __restrict__ sbuf,    // [O][B][D]
    float* __restrict__ logits)        // [O][C][B]
{
    const int tid   = threadIdx.x;
    const int lane  = tid & 31;
    const int wave  = tid >> 5;
    const int bb    = blockIdx.y * 16;
    const int bl    = lane & 15;
    const int hi    = lane >> 4;
    const int cbase = blockIdx.x * 144 + wave * 18;   // 18 c's (9 pairs) / wave
    const int n     = lane & 15;       // N col of tile -> (c within pair, i)
    const int ii    = n & 7;
    const int csel  = n >> 3;

    for (int o = 0; o < O_CAPS; ++o) {
        // A = s[b, d] : 4 K-steps of 4
        const float* sp = sbuf + (o * B_TOT + bb + bl) * D_DIM;
        v2f a0 = *(const v2f*)(sp + 0  + hi * 2);
        v2f a1 = *(const v2f*)(sp + 4  + hi * 2);
        v2f a2 = *(const v2f*)(sp + 8  + hi * 2);
        v2f a3 = *(const v2f*)(sp + 12 + hi * 2);

        for (int p = 0; p < 9; ++p) {
            int cc = cbase + p * 2 + csel;
            const float* wp = W + (size_t)(o * C_IN + cc) * D_DIM * I_IN
                                + hi * 2 * I_IN + ii;
            v8f g = {};
#pragma unroll
            for (int k0 = 0; k0 < 16; k0 += 4) {
                v2f bm;
                bm.x = wp[(k0    ) * I_IN];
                bm.y = wp[(k0 + 1) * I_IN];
                v2f a = (k0 == 0) ? a0 : (k0 == 4) ? a1 : (k0 == 8) ? a2 : a3;
                g = __builtin_amdgcn_wmma_f32_16x16x4_f32(false, a, false, bm,
                                                          (short)0, g, false, false);
            }
            // uv epilogue: dot with x over i (8 lanes per group)
#pragma unroll
            for (int j = 0; j < 8; ++j) {
                int bg = bb + ((lane < 16) ? j : j + 8);
                float prod = g[j] * x[(bg * C_IN + cc) * I_IN + ii];
                prod += __shfl_xor(prod, 1);
                prod += __shfl_xor(prod, 2);
                prod += __shfl_xor(prod, 4);
                if (ii == 0) {
                    int idx = (o * C_IN + cc) * B_TOT + bg;
                    logits[idx] = ACC ? (logits[idx] + prod) : prod;
                }
            }
        }
    }
}

// ---------------------------------------------------------------------------
// coeff[o,c,b] = softmax over o of logits[o,c,b]   (one thread per (c,b))
// ---------------------------------------------------------------------------
__global__ __launch_bounds__(256) void softmax_kernel(
    const float* __restrict__ logits, float* __restrict__ coeff)
{
    int t = blockIdx.x * 256 + threadIdx.x;   // [0, C*B)
    int b = t & (B_TOT - 1);
    int c = t >> 9;
    float l[O_CAPS];
    float m = -1e30f;
#pragma unroll
    for (int o = 0; o < O_CAPS; ++o) {
        l[o] = logits[(o * C_IN + c) * B_TOT + b];
        m = fmaxf(m, l[o]);
    }
    float s = 0.f;
#pragma unroll
    for (int o = 0; o < O_CAPS; ++o) { l[o] = __expf(l[o] - m); s += l[o]; }
    float inv = 1.f / s;
#pragma unroll
    for (int o = 0; o < O_CAPS; ++o)
        coeff[(o * C_IN + c) * B_TOT + b] = l[o] * inv;
}

// ---------------------------------------------------------------------------
extern "C" void kernel_launch(void* const* d_in, const int* in_sizes, int n_in,
                              void* d_out, int out_size, void* d_ws, size_t ws_size,
                              hipStream_t stream) {
    const float* x = (const float*)d_in[0];   // [512][1152][8]
    const float* W = (const float*)d_in[1];   // [1][10][1152][16][8]
    float* out = (float*)d_out;               // [512][10][16]

    float* ws     = (float*)d_ws;
    float* xT     = ws;                                   // 4,718,592 f
    float* logits = xT + (size_t)C_IN * I_IN * B_TOT;     // 5,898,240 f
    float* coeff  = logits + (size_t)O_CAPS * C_IN * B_TOT;
    float* sbuf   = coeff + (size_t)O_CAPS * C_IN * B_TOT; // 81,920 f

    const dim3 gs(O_CAPS, B_TOT / 16);   // (o, batch-tile)
    const dim3 gu(8, B_TOT / 16);        // (c-chunk, batch-tile)

    transpose_x_kernel<<<(B_TOT * C_IN * I_IN) / 256, 256, 0, stream>>>(x, xT);

    // iter 1: coeff = softmax(0) = 0.1
    gemm_s_kernel<0><<<gs, 256, 0, stream>>>(xT, W, coeff, sbuf);
    gemm_uv_kernel<false><<<gu, 256, 0, stream>>>(x, W, sbuf, logits); // b1 = uv1
    softmax_kernel<<<(C_IN * B_TOT) / 256, 256, 0, stream>>>(logits, coeff);

    // iter 2
    gemm_s_kernel<1><<<gs, 256, 0, stream>>>(xT, W, coeff, sbuf);
    gemm_uv_kernel<true><<<gu, 256, 0, stream>>>(x, W, sbuf, logits); // b2 = b1+uv2
    softmax_kernel<<<(C_IN * B_TOT) / 256, 256, 0, stream>>>(logits, coeff);

    // final: v = s3, reference layout [B][O][D]
    gemm_s_kernel<2><<<gs, 256, 0, stream>>>(xT, W, coeff, out);
}